// LocallyConnected2d_16501264351337
// MI455X (gfx1250) — compile-verified
//
#include <hip/hip_runtime.h>
#include <hip/hip_bf16.h>
#include <math.h>

// Problem constants
#define B_    16
#define CIN   32
#define H_    64
#define W_    64
#define COUT  64
#define HOUT  62
#define WOUT  62
#define KH_   3
#define KW_   3
#define KDIM  (CIN * KH_ * KW_)   // 288

typedef __attribute__((ext_vector_type(2))) float v2f;
typedef __attribute__((ext_vector_type(8))) float v8f;

// One 128-thread workgroup (4 waves) per output spatial position (i,j).
// Wave w computes the 16(batch) x 16(cout tile w) GEMM tile over K=288
// using v_wmma_f32_16x16x4_f32, accumulating in 8 VGPRs.
__global__ __launch_bounds__(128) void lc2d_wmma_kernel(
    const float* __restrict__ x,      // [16, 32, 64, 64]
    const float* __restrict__ wgt,    // [62, 62, 64, 32, 3, 3]
    float* __restrict__ out)          // [16, 64, 62, 62]
{
    // A matrix staged in LDS, layout A[k][b] (k-major) so the WMMA A-fragment
    // read (16 consecutive b per half-wave) is bank-conflict free.
    __shared__ float a_lds[KDIM * B_];     // 18 KB

    const int pos  = blockIdx.x;           // 0 .. 3843
    const int i    = pos / WOUT;
    const int j    = pos - i * WOUT;
    const int tid  = threadIdx.x;          // 0 .. 127
    const int lane = tid & 31;
    const int wave = tid >> 5;             // n-tile index, 0..3

    // ---- Stage x patch into LDS: A[k][b] = x[b, c, i+kh, j+kw], k=c*9+kh*3+kw
    {
        const float* xb = x + i * W_ + j;
        #pragma unroll
        for (int e = tid; e < KDIM * B_; e += 128) {
            const int k  = e >> 4;         // 0..287
            const int b  = e & 15;         // 0..15
            const int c  = k / 9;
            const int r  = k - c * 9;
            const int kh = r / 3;
            const int kw = r - kh * 3;
            a_lds[k * B_ + b] =
                xb[b * (CIN * H_ * W_) + c * (H_ * W_) + kh * W_ + kw];
        }
    }
    __syncthreads();

    // ---- Per-lane bases for WMMA fragments
    const int m     = lane & 15;               // batch row of this lane
    const int kLane = (lane >> 4) << 1;        // 0 (lanes 0-15) or 2 (lanes 16-31)
    const int n     = wave * 16 + m;           // output channel of this lane

    // Weight block for this position is contiguous: [COUT][KDIM]
    const float* __restrict__ wrow =
        wgt + (size_t)pos * (COUT * KDIM) + (size_t)n * KDIM + kLane;

    v8f acc = {0.f, 0.f, 0.f, 0.f, 0.f, 0.f, 0.f, 0.f};

    // ---- K loop: 72 steps of K=4. B frag = one global b64 per lane (streaming
    // the 283MB weight tensor); A frag = one conflict-free 2-address LDS read.
    #pragma unroll 8
    for (int k0 = 0; k0 < KDIM; k0 += 4) {
        v2f afrag;
        afrag.x = a_lds[(k0 + kLane) * B_ + m];
        afrag.y = a_lds[(k0 + kLane + 1) * B_ + m];

        v2f bfrag;
        bfrag.x = wrow[k0];        // contiguous pair -> global_load_b64
        bfrag.y = wrow[k0 + 1];

        acc = __builtin_amdgcn_wmma_f32_16x16x4_f32(
            /*neg_a=*/false, afrag,
            /*neg_b=*/false, bfrag,
            /*c_mod=*/(short)0, acc,
            /*reuse_a=*/false, /*reuse_b=*/false);
    }

    // ---- Epilogue: sigmoid (fast: v_exp_f32 + v_rcp_f32) + store.
    // D layout: VGPR r holds batch row (r + 8*(lane>=16)), col n = lane&15.
    const int  mbase = (lane >> 4) << 3;                 // 0 or 8
    float* __restrict__ op =
        out + (size_t)n * (HOUT * WOUT) + (size_t)i * WOUT + j;
    #pragma unroll
    for (int r2 = 0; r2 < 8; ++r2) {
        const float v = acc[r2];
        const float s = __builtin_amdgcn_rcpf(1.0f + __expf(-v));
        op[(size_t)(mbase + r2) * (COUT * HOUT * WOUT)] = s;
    }
}

extern "C" void kernel_launch(void* const* d_in, const int* in_sizes, int n_in,
                              void* d_out, int out_size, void* d_ws, size_t ws_size,
                              hipStream_t stream) {
    const float* x   = (const float*)d_in[0];   // 16*32*64*64
    const float* wgt = (const float*)d_in[1];   // 62*62*64*32*3*3
    float* out = (float*)d_out;                 // 16*64*62*62

    dim3 grid(HOUT * WOUT);   // 3844 workgroups, one per spatial position
    dim3 block(128);          // 4 waves: one 16-wide cout tile each
    hipLaunchKernelGGL(lc2d_wmma_kernel, grid, block, 0, stream, x, wgt, out);
}